// SSIM_1846835938197
// MI455X (gfx1250) — compile-verified
//
#include <hip/hip_runtime.h>
#include <math.h>

// Problem geometry (32,3,384,512) fp32
#define HH      384
#define WW      512
#define NPLANES 96          // N*C
#define TILE_R  64          // output rows per workgroup
#define TILE_C  32          // output cols per workgroup
#define RADIUS  5

// LDS layout (floats). Pitch 44 is bank-conflict-free for all access patterns:
//  - A b64 loads: row*44 mod 64 distinct for 16 rows, pairs disjoint across halves
//  - C stores: half-wave rows differ by 8 -> 352 mod 64 = 32 (disjoint banks)
//  - vertical B loads: rows differ by 2 -> 88 mod 64 = 24 (disjoint banks)
#define PITCH     44
#define IMG_ROWS  80        // 5 chunks of 16 rows (rows >= 74 zero-filled)
#define LOAD_ROWS 74        // 64 + 2*RADIUS real halo rows
#define LOAD_COLS 42        // 32 + 2*RADIUS real halo cols
#define S1OFF 0
#define S2OFF (IMG_ROWS*PITCH)            // 3520
#define QSZ   (IMG_ROWS*PITCH)            // 3520 per h-buffer
#define HOFF  (2*QSZ)                     // 7040
#define REDOFF (HOFF + 5*QSZ)             // 24640
#define SMEM_FLOATS (REDOFF + 1)          // 24641 floats ~= 98.6 KB

typedef __attribute__((ext_vector_type(2))) float v2f;
typedef __attribute__((ext_vector_type(8))) float v8f;

// D = A(16x4 f32) x B(4x16 f32) + C(16x16 f32), full precision on the matrix pipe
__device__ __forceinline__ v8f wmma4(v2f a, v2f b, v8f c) {
  return __builtin_amdgcn_wmma_f32_16x16x4_f32(
      /*neg_a=*/false, a, /*neg_b=*/false, b,
      /*c_mod=*/(short)0, c, /*reuse_a=*/false, /*reuse_b=*/false);
}

// Banded Gaussian: g_band(i) = phi[i] for i in [0,10], else 0.
// phi[t] = exp(-(t-5)^2/(2*sigma^2)) / sum, sigma = 1.5 (matches reference)
__device__ __forceinline__ float gband(int i, float invS) {
  if (i < 0 || i > 10) return 0.0f;
  float d = (float)(i - 5);
  return expf(d * d * (-1.0f / 4.5f)) * invS;
}

// One 16x16 horizontal-conv tile for quantity Q (compile-time): product
// selection and h-buffer offset fold to constants; body is straight-line
// (14 LDS loads + 7 WMMAs in two independent accumulator chains).
template <int Q>
__device__ __forceinline__ void stage2_tile(float* sm, int rem, int tlo,
                                            int khi, int lane, const v2f* Wc) {
  const int r0 = (rem >> 1) << 4;       // {0,16,32,48,64}  (scalar)
  const int cb = (rem & 1) << 4;        // {0,16}           (scalar)
  v8f acc0 = {0.f, 0.f, 0.f, 0.f, 0.f, 0.f, 0.f, 0.f};
  v8f acc1 = {0.f, 0.f, 0.f, 0.f, 0.f, 0.f, 0.f, 0.f};
#pragma unroll
  for (int c = 0; c < 7; ++c) {
    int ao = (r0 + tlo) * PITCH + (cb + 4 * c + khi);  // 8B-aligned (even col)
    v2f A;
    if constexpr (Q == 0) {
      A = *(const v2f*)(sm + S1OFF + ao);
    } else if constexpr (Q == 1) {
      A = *(const v2f*)(sm + S2OFF + ao);
    } else if constexpr (Q == 2) {
      v2f a1 = *(const v2f*)(sm + S1OFF + ao);
      A = a1 * a1;
    } else if constexpr (Q == 3) {
      v2f a2 = *(const v2f*)(sm + S2OFF + ao);
      A = a2 * a2;
    } else {
      v2f a1 = *(const v2f*)(sm + S1OFF + ao);
      v2f a2 = *(const v2f*)(sm + S2OFF + ao);
      A = a1 * a2;
    }
    if (c & 1) acc1 = wmma4(A, Wc[c], acc1);
    else       acc0 = wmma4(A, Wc[c], acc0);
  }
  v8f acc = acc0 + acc1;
  float* hq = sm + HOFF + Q * QSZ;      // compile-time offset
  int col = cb + tlo;
  int rb  = r0 + ((lane < 16) ? 0 : 8);
#pragma unroll
  for (int vg = 0; vg < 8; ++vg) hq[(rb + vg) * PITCH + col] = acc[vg];
}

__global__ void __launch_bounds__(256)
ssim_tile_kernel(const float* __restrict__ img1, const float* __restrict__ img2,
                 float* __restrict__ partial) {
  extern __shared__ float sm[];
  const int tid  = threadIdx.x;
  const int lane = tid & 31;
  // Force the wave index into an SGPR so all control flow derived from it is
  // scalar (s_cbranch) -> EXEC stays all-ones around every WMMA (ISA req).
  const int wsc  = __builtin_amdgcn_readfirstlane(tid >> 5);
  const int tlo  = lane & 15;               // M (A) / N (B,C) index of this lane
  const int khi  = (lane < 16) ? 0 : 2;     // K base held by this half-wave

  const int tileX = blockIdx.x;             // 16 tiles of 32 cols
  const int tileY = blockIdx.y;             // 6 tiles of 64 rows
  const int plane = blockIdx.z;             // 96 planes
  const long planeBase = (long)plane * (HH * WW);
  const int gr0 = tileY * TILE_R - RADIUS;
  const int gc0 = tileX * TILE_C - RADIUS;

  if (tid == 0) sm[REDOFF] = 0.0f;

  // ---- per-lane banded-filter constants (serve as B in stage 2, A in stage 3)
  float invS;
  {
    float s = 1.0f;
#pragma unroll
    for (int d = 1; d <= 5; ++d) s += 2.0f * expf(-(float)(d * d) * (1.0f / 4.5f));
    invS = 1.0f / s;
  }
  v2f Wc[7];
#pragma unroll
  for (int c = 0; c < 7; ++c) {
    int i0 = 4 * c + khi - tlo;
    Wc[c].x = gband(i0, invS);
    Wc[c].y = gband(i0 + 1, invS);
  }

  // ---- Stage 1: cooperative halo load, uniform trip count (14 = 3584/256),
  // zero-padded (cols 42..43 / rows 74..79 zeroed; borders zero-padded).
#pragma unroll 1
  for (int base = 0; base < 14 * 256; base += 256) {
    int idx = base + tid;
    if (idx < IMG_ROWS * PITCH) {
      int r = idx / PITCH;
      int c = idx - r * PITCH;
      float a = 0.0f, b = 0.0f;
      if (r < LOAD_ROWS && c < LOAD_COLS) {
        int gr = gr0 + r, gc = gc0 + c;
        if (gr >= 0 && gr < HH && gc >= 0 && gc < WW) {
          long off = planeBase + (long)gr * WW + gc;
          a = img1[off];
          b = img2[off];
        }
      }
      sm[S1OFF + idx] = a;
      sm[S2OFF + idx] = b;
    }
  }
  __syncthreads();

  // ---- Stage 2: horizontal conv via WMMA for 5 quantities {x, y, x^2, y^2, xy}.
  // Per q: 10 tiles (5 row-chunks x 2 col-chunks). Every wave does tile `wsc`;
  // tiles 8,9 go to a q-rotated pair of waves so totals stay balanced (7 vs 6).
#define STAGE2_Q(Q)                                                 \
  {                                                                 \
    stage2_tile<Q>(sm, wsc, tlo, khi, lane, Wc);                    \
    int h = (wsc - 2 * (Q)) & 7; /* scalar */                       \
    if (h < 2) stage2_tile<Q>(sm, 8 + h, tlo, khi, lane, Wc);       \
  }
  STAGE2_Q(0)
  STAGE2_Q(1)
  STAGE2_Q(2)
  STAGE2_Q(3)
  STAGE2_Q(4)
#undef STAGE2_Q
  __syncthreads();

  // ---- Stage 3: vertical conv via WMMA; each wave owns one 16x16 output tile,
  // computes all 5 filtered quantities in registers.
  const int m0  = (wsc >> 1) << 4;   // {0,16,32,48}  (scalar)
  const int cb3 = (wsc & 1) << 4;    // {0,16}        (scalar)
  v8f vq[5];
#pragma unroll
  for (int q = 0; q < 5; ++q) {
    v8f acc0 = {0.f, 0.f, 0.f, 0.f, 0.f, 0.f, 0.f, 0.f};
    v8f acc1 = {0.f, 0.f, 0.f, 0.f, 0.f, 0.f, 0.f, 0.f};
    const float* hq = sm + HOFF + q * QSZ;
#pragma unroll
    for (int c = 0; c < 7; ++c) {
      int kr = m0 + 4 * c + khi;
      int co = cb3 + tlo;
      v2f B;
      B.x = hq[kr * PITCH + co];
      B.y = hq[(kr + 1) * PITCH + co];
      if (c & 1) acc1 = wmma4(Wc[c], B, acc1);   // A is the banded constant here
      else       acc0 = wmma4(Wc[c], B, acc0);
    }
    vq[q] = acc0 + acc1;
  }

  // ---- Pointwise SSIM + reduction
  const float C1c = 0.0001f;  // 0.01^2
  const float C2c = 0.0009f;  // 0.03^2
  float ssum = 0.0f;
#pragma unroll
  for (int vg = 0; vg < 8; ++vg) {
    float mx  = vq[0][vg], my  = vq[1][vg];
    float ex2 = vq[2][vg], ey2 = vq[3][vg], exy = vq[4][vg];
    float mx2 = mx * mx, my2 = my * my, mxy = mx * my;
    float vx = ex2 - mx2, vy = ey2 - my2, cv = exy - mxy;
    float num = (2.0f * mxy + C1c) * (2.0f * cv + C2c);
    float den = (mx2 + my2 + C1c) * (vx + vy + C2c);
    ssum += num / den;
  }
#pragma unroll
  for (int off = 16; off > 0; off >>= 1) ssum += __shfl_xor(ssum, off, 32);
  if (lane == 0) atomicAdd(&sm[REDOFF], ssum);
  __syncthreads();
  if (tid == 0) {
    int bid = (blockIdx.z * gridDim.y + blockIdx.y) * gridDim.x + blockIdx.x;
    partial[bid] = sm[REDOFF];
  }
}

__global__ void __launch_bounds__(256)
ssim_reduce_kernel(const float* __restrict__ partial, int n,
                   float* __restrict__ out, float scale) {
  __shared__ float acc;
  if (threadIdx.x == 0) acc = 0.0f;
  __syncthreads();
  float s = 0.0f;
  for (int i = threadIdx.x; i < n; i += 256) s += partial[i];
#pragma unroll
  for (int off = 16; off > 0; off >>= 1) s += __shfl_xor(s, off, 32);
  if ((threadIdx.x & 31) == 0) atomicAdd(&acc, s);
  __syncthreads();
  if (threadIdx.x == 0) out[0] = acc * scale;
}

extern "C" void kernel_launch(void* const* d_in, const int* in_sizes, int n_in,
                              void* d_out, int out_size, void* d_ws, size_t ws_size,
                              hipStream_t stream) {
  const float* img1 = (const float*)d_in[0];
  const float* img2 = (const float*)d_in[1];
  float* out  = (float*)d_out;
  float* part = (float*)d_ws;            // 9216 floats = 36 KB of scratch

  dim3 grid(WW / TILE_C, HH / TILE_R, NPLANES);   // (16, 6, 96)
  size_t smemBytes = (size_t)SMEM_FLOATS * sizeof(float);
  ssim_tile_kernel<<<grid, 256, smemBytes, stream>>>(img1, img2, part);

  const int nPart = (WW / TILE_C) * (HH / TILE_R) * NPLANES;  // 9216
  const float scale = 1.0f / (float)((long)NPLANES * HH * WW); // 1/18874368
  ssim_reduce_kernel<<<1, 256, 0, stream>>>(part, nPart, out, scale);
}